// ShardedEmbedIn_20083267076366
// MI455X (gfx1250) — compile-verified
//
#include <hip/hip_runtime.h>
#include <stdint.h>

// ShardedEmbedIn on MI455X (gfx1250).
//
// Reference math collapses to a row gather: one_hot(id - s*dps) @ W[s] summed
// over shards == W_flat[id, :], because shard*dps + within == id for the
// stacked [TP, dps, H] table. So this is pure data movement:
//   8192 rows x 6144 fp16 = ~100.7 MB read + ~100.7 MB write
//   => ~8.6 us at 23.3 TB/s HBM. Zero FLOPs; WMMA would be ~6000x more work.
// We use the CDNA5 async Global<->LDS DMA path (ASYNCcnt) to stream rows.

#define HIDDEN      6144
#define ROW_BYTES   (HIDDEN * 2)              // 12288 bytes per fp16 row
#define BLOCK       256                       // 8 wave32s per workgroup
#define CHUNK_BYTES (BLOCK * 16)              // 4096 B per block-wide B128 op
#define NCHUNK      (ROW_BYTES / CHUNK_BYTES) // 3

// The async-LDS builtins take pointers to 16-byte int vectors:
//   arg0: v4i32 in the global (AS1 / __device__) address space
//   arg1: v4i32 in the LDS (AS3 / __shared__) address space
typedef int v4i __attribute__((ext_vector_type(4)));
typedef __attribute__((address_space(1))) v4i gv4i;
typedef __attribute__((address_space(3))) v4i lv4i;

__device__ __forceinline__ void async_load_b128(const void* gsrc, void* lds) {
#if __has_builtin(__builtin_amdgcn_global_load_async_to_lds_b128)
    __builtin_amdgcn_global_load_async_to_lds_b128(
        (gv4i*)(void*)gsrc, (lv4i*)lds, /*offset=*/0, /*cpol=*/0);
#else
    unsigned lds_addr = (unsigned)(uintptr_t)(lv4i*)lds;
    asm volatile("global_load_async_to_lds_b128 %0, %1, off"
                 :
                 : "v"(lds_addr), "v"(gsrc)
                 : "memory");
#endif
}

__device__ __forceinline__ void async_store_b128(void* gdst, const void* lds) {
#if __has_builtin(__builtin_amdgcn_global_store_async_from_lds_b128)
    __builtin_amdgcn_global_store_async_from_lds_b128(
        (gv4i*)gdst, (lv4i*)(void*)lds, /*offset=*/0, /*cpol=*/0);
#else
    unsigned lds_addr = (unsigned)(uintptr_t)(lv4i*)(void*)lds;
    asm volatile("global_store_async_from_lds_b128 %0, %1, off"
                 :
                 : "v"(gdst), "v"(lds_addr)
                 : "memory");
#endif
}

__device__ __forceinline__ void wait_asynccnt0() {
#if __has_builtin(__builtin_amdgcn_s_wait_asynccnt)
    __builtin_amdgcn_s_wait_asynccnt(0);
#else
    asm volatile("s_wait_asynccnt 0" ::: "memory");
#endif
}

__global__ __launch_bounds__(BLOCK) void ShardedEmbedIn_gather_kernel(
    const int* __restrict__ input_ids,     // [B*S], values in [0, VOCAB)
    const _Float16* __restrict__ W,        // [TP * dps * HIDDEN] fp16, row id = token id
    _Float16* __restrict__ out)            // [B*S, HIDDEN] fp16
{
    __shared__ __align__(16) char row_buf[ROW_BYTES];

    const int token = (int)blockIdx.x;
    const unsigned id = (unsigned)input_ids[token];   // uniform per block -> scalar load

    const char* src = (const char*)W + (size_t)id * (size_t)ROW_BYTES;
    char*       dst = (char*)out + (size_t)token * (size_t)ROW_BYTES;
    const int lane_off = (int)threadIdx.x * 16;       // 16 B per lane per op

    // Stream the whole 12 KB row into LDS with async B128 transfers.
    // Each wave issues ops covering exactly its own lanes' bytes, so the
    // per-wave ASYNCcnt wait below is a complete load->store dependency.
#pragma unroll
    for (int c = 0; c < NCHUNK; ++c) {
        const int off = lane_off + c * CHUNK_BYTES;
        async_load_b128(src + off, row_buf + off);
    }

    wait_asynccnt0();   // s_wait_asynccnt 0: LDS now holds this wave's bytes

#pragma unroll
    for (int c = 0; c < NCHUNK; ++c) {
        const int off = lane_off + c * CHUNK_BYTES;
        async_store_b128(dst + off, row_buf + off);
    }
    // Outstanding async stores drain before S_ENDPGM's implicit wait-idle.
}

extern "C" void kernel_launch(void* const* d_in, const int* in_sizes, int n_in,
                              void* d_out, int out_size, void* d_ws, size_t ws_size,
                              hipStream_t stream) {
    const int*      input_ids = (const int*)d_in[0];       // [B*S] token ids
    const _Float16* W         = (const _Float16*)d_in[1];  // [TP, dps, HIDDEN] fp16
    _Float16*       out       = (_Float16*)d_out;          // [B*S, HIDDEN] fp16

    const int n_tokens = in_sizes[0];   // BATCH * SEQ = 8192

    ShardedEmbedIn_gather_kernel<<<dim3((unsigned)n_tokens), dim3(BLOCK), 0, stream>>>(
        input_ids, W, out);

    (void)n_in; (void)out_size; (void)d_ws; (void)ws_size;
}